// Get_Logit_MC_66503273611873
// MI455X (gfx1250) — compile-verified
//
#include <hip/hip_runtime.h>

// CDNA5 / gfx1250: wave32, WMMA (not MFMA).
// One wave32 computes one batch element's full 16x16 pairwise-distance tile
// via V_WMMA_F32_16X16X4_F32 accumulation over K=64 in 16 steps of K=4.

typedef __attribute__((ext_vector_type(2))) float v2f;
typedef __attribute__((ext_vector_type(8))) float v8f;

#define S 16
#define D 64
#define WAVES_PER_BLOCK 8

__global__ __launch_bounds__(256) void get_logit_mc_kernel(
    const float* __restrict__ z1,
    const float* __restrict__ z2,
    const float* __restrict__ a_ptr,
    const float* __restrict__ b_ptr,
    float* __restrict__ out,
    int B)
{
    const int lane = threadIdx.x & 31;
    const int wave = threadIdx.x >> 5;
    const int batch = blockIdx.x * WAVES_PER_BLOCK + wave;
    if (batch >= B) return;   // wave-uniform: EXEC stays all-ones for WMMA

    const float a  = a_ptr[0];
    const float bb = b_ptr[0];

    const int half = lane >> 4;   // 0: lanes 0-15, 1: lanes 16-31
    const int row  = lane & 15;   // m-index for z1 (A), n-index for z2 (B)

    // A-matrix 16x4 fp32 layout: lane 16*half+m holds K = k0+2*half+{0,1}
    // B-matrix 4x16 fp32 layout: lane 16*half+n holds K = k0+2*half+{0,1}
    // -> identical per-lane addressing, z1 vs z2.
    const float* z1b = z1 + (size_t)batch * (S * D) + (size_t)row * D + 2 * half;
    const float* z2b = z2 + (size_t)batch * (S * D) + (size_t)row * D + 2 * half;

    v8f acc = {0.f, 0.f, 0.f, 0.f, 0.f, 0.f, 0.f, 0.f};
    float sq1p = 0.f;   // partial sum of squares of the A elements this lane holds
    float sq2p = 0.f;   // partial sum of squares of the B elements this lane holds

#pragma unroll
    for (int k0 = 0; k0 < D; k0 += 4) {
        v2f av = *(const v2f*)(z1b + k0);
        v2f bv = *(const v2f*)(z2b + k0);
        sq1p += av.x * av.x + av.y * av.y;
        sq2p += bv.x * bv.x + bv.y * bv.y;
        // D = A x B + C, fp32 in / fp32 accumulate
        acc = __builtin_amdgcn_wmma_f32_16x16x4_f32(
            /*neg_a=*/false, av,
            /*neg_b=*/false, bv,
            /*c_mod=*/(short)0, acc,
            /*reuse_a=*/false, /*reuse_b=*/false);
    }

    // Complete the row norms: lane (16*half + r) holds half of row r's sum.
    sq1p += __shfl_xor(sq1p, 16, 32);   // now lane L holds sq1[L & 15]
    sq2p += __shfl_xor(sq2p, 16, 32);   // now lane L holds sq2[L & 15]

    // C/D tile layout: VGPR v -> M = v + 8*half, N = row.
    // sq2[N] is this lane's own sq2p (N == row). sq1[M] needs a broadcast.
    float* outb = out + (size_t)batch * (S * S);

#pragma unroll
    for (int v = 0; v < 8; ++v) {
        const int M = v + 8 * half;
        const float dot = acc[v];
        const float s1  = __shfl(sq1p, M, 32);   // sq1[M] lives in lane M
        float d2 = s1 + sq2p - 2.0f * dot;
        float dist = __builtin_sqrtf(fmaxf(d2, 0.0f));
        float x = dist * a + bb;
        float logit = 1.0f / (1.0f + __expf(-x));
        logit = fminf(fmaxf(logit, 1e-8f), 1.0f);
        outb[M * S + row] = logit;
    }
}

extern "C" void kernel_launch(void* const* d_in, const int* in_sizes, int n_in,
                              void* d_out, int out_size, void* d_ws, size_t ws_size,
                              hipStream_t stream) {
    (void)n_in; (void)out_size; (void)d_ws; (void)ws_size;

    const float* z1 = (const float*)d_in[0];
    const float* z2 = (const float*)d_in[1];
    const float* a  = (const float*)d_in[2];   // scalar (in_sizes[2]==1)
    const float* b  = (const float*)d_in[3];   // scalar (in_sizes[3]==1)
    float* out = (float*)d_out;

    const int B = in_sizes[0] / (S * D);       // 32768 for the reference shapes
    const int blocks = (B + WAVES_PER_BLOCK - 1) / WAVES_PER_BLOCK;

    hipLaunchKernelGGL(get_logit_mc_kernel,
                       dim3(blocks), dim3(32 * WAVES_PER_BLOCK), 0, stream,
                       z1, z2, a, b, out, B);
}